// GnnBlock_33225867002466
// MI455X (gfx1250) — compile-verified
//
#include <hip/hip_runtime.h>

typedef __attribute__((ext_vector_type(2))) float v2f;
typedef __attribute__((ext_vector_type(8))) float v8f;

// ---------------------------------------------------------------- utilities
__global__ void zero_f32_kernel(float* __restrict__ p, int n) {
    int i = blockIdx.x * blockDim.x + threadIdx.x;
    if (i < n) p[i] = 0.0f;
}

__global__ void copy_i32_kernel(const int* __restrict__ in, int* __restrict__ out, int n) {
    int i = blockIdx.x * blockDim.x + threadIdx.x;
    if (i < n) out[i] = in[i];
}

// ------------------------------------------------------------ degree counts
__global__ void degree_kernel(const int* __restrict__ dst, float* __restrict__ cnt, int E) {
    int e = blockIdx.x * blockDim.x + threadIdx.x;
    if (e < E) atomicAdd(&cnt[dst[e]], 1.0f);
}

// --------------------------------------------------- edge scatter (gather+add)
// thread = (edge, 4-feature chunk): float4 gather from x[src], 4 atomic adds.
__global__ void scatter_kernel(const float* __restrict__ x,
                               const int* __restrict__ src,
                               const int* __restrict__ dst,
                               float* __restrict__ agg, int E) {
    int idx = blockIdx.x * blockDim.x + threadIdx.x;
    if (idx >= E * 16) return;
    int e = idx >> 4;
    int q = idx & 15;
    int s = src[e];
    int d = dst[e];
    const float4 v = *(const float4*)(x + (size_t)s * 64 + q * 4);
    float* p = agg + (size_t)d * 64 + q * 4;
    atomicAdd(p + 0, v.x);
    atomicAdd(p + 1, v.y);
    atomicAdd(p + 2, v.z);
    atomicAdd(p + 3, v.w);
}

// ------------------------------------------------- fused SAGE layer (WMMA f32)
// out = leaky_relu( [agg/deg | x] @ [Wl^T ; Wr^T] + bl ) + x
// One wave -> 16 rows x 64 cols. K = 128 as 32 steps of V_WMMA_F32_16X16X4_F32.
__global__ __launch_bounds__(256) void sage_layer_kernel(
    const float* __restrict__ x_in, const float* __restrict__ aggsum,
    const float* __restrict__ cnt,
    const float* __restrict__ Wl, const float* __restrict__ bl,
    const float* __restrict__ Wr,
    float* __restrict__ x_out, int N)
{
    __shared__ float Wsh[128 * 64];   // B matrix: Wsh[k*64 + j] = W^T combined
    __shared__ float bsh[64];

    const int tid = threadIdx.x;
    for (int i = tid; i < 128 * 64; i += 256) {
        const int k = i >> 6, j = i & 63;
        Wsh[i] = (k < 64) ? Wl[(size_t)j * 64 + k] : Wr[(size_t)j * 64 + (k - 64)];
    }
    if (tid < 64) bsh[tid] = bl[tid];
    __syncthreads();

    const int lane = tid & 31;
    const int wave = tid >> 5;
    const int m0   = (blockIdx.x * 8 + wave) * 16;
    if (m0 >= N) return;                      // wave-uniform exit, EXEC stays full

    const int col  = lane & 15;               // N position / A row within tile
    const int koff = (lane >> 4) << 1;        // lanes 16-31 hold K+2,K+3
    const int rA   = m0 + col;                // A-matrix row owned by this lane
    const int rc   = (rA < N) ? rA : (N - 1); // clamp for safe loads

    const float inv = 1.0f / fmaxf(cnt[rc], 1.0f);
    const float* __restrict__ arow = aggsum + (size_t)rc * 64;
    const float* __restrict__ xrow = x_in   + (size_t)rc * 64;

    v8f acc0 = {}, acc1 = {}, acc2 = {}, acc3 = {};

#define WMMA_STEP(K0, AV)                                                           \
    {                                                                               \
        const float* wr0 = &Wsh[(K0) * 64 + col];                                   \
        const float* wr1 = wr0 + 64;                                                \
        v2f b0, b1, b2, b3;                                                         \
        b0.x = wr0[0];  b0.y = wr1[0];                                              \
        b1.x = wr0[16]; b1.y = wr1[16];                                             \
        b2.x = wr0[32]; b2.y = wr1[32];                                             \
        b3.x = wr0[48]; b3.y = wr1[48];                                             \
        acc0 = __builtin_amdgcn_wmma_f32_16x16x4_f32(false, (AV), false, b0, (short)0, acc0, false, false); \
        acc1 = __builtin_amdgcn_wmma_f32_16x16x4_f32(false, (AV), false, b1, (short)0, acc1, false, false); \
        acc2 = __builtin_amdgcn_wmma_f32_16x16x4_f32(false, (AV), false, b2, (short)0, acc2, false, false); \
        acc3 = __builtin_amdgcn_wmma_f32_16x16x4_f32(false, (AV), false, b3, (short)0, acc3, false, false); \
    }

    // K = 0..63 : neighbor-mean half of A
#pragma unroll
    for (int kb = 0; kb < 16; ++kb) {
        const int k0 = kb * 4 + koff;
        v2f a;
        a.x = arow[k0] * inv;
        a.y = arow[k0 + 1] * inv;
        WMMA_STEP(k0, a);
    }
    // K = 64..127 : self half of A
#pragma unroll
    for (int kb = 0; kb < 16; ++kb) {
        const int k0 = kb * 4 + koff;
        v2f a;
        a.x = xrow[k0];
        a.y = xrow[k0 + 1];
        WMMA_STEP(k0 + 64, a);
    }
#undef WMMA_STEP

    // Epilogue: C/D layout -> lane owns col (lane&15), rows rbase..rbase+7.
    const int rbase = m0 + ((lane >> 4) << 3);
#define EPILOGUE(ACC, NT)                                                           \
    {                                                                               \
        const int   c  = (NT) * 16 + col;                                           \
        const float bb = bsh[c];                                                    \
        _Pragma("unroll")                                                           \
        for (int i = 0; i < 8; ++i) {                                               \
            const int r = rbase + i;                                                \
            if (r < N) {                                                            \
                float h = ACC[i] + bb;                                              \
                h = (h >= 0.0f) ? h : 0.01f * h;                                    \
                x_out[(size_t)r * 64 + c] = h + x_in[(size_t)r * 64 + c];           \
            }                                                                       \
        }                                                                           \
    }
    EPILOGUE(acc0, 0)
    EPILOGUE(acc1, 1)
    EPILOGUE(acc2, 2)
    EPILOGUE(acc3, 3)
#undef EPILOGUE
}

// ------------------------------------------------------------------- launcher
extern "C" void kernel_launch(void* const* d_in, const int* in_sizes, int n_in,
                              void* d_out, int out_size, void* d_ws, size_t ws_size,
                              hipStream_t stream) {
    const float* x     = (const float*)d_in[0];
    const int*   ei    = (const int*)d_in[1];
    const int*   batch = (const int*)d_in[2];
    const float* Wl1   = (const float*)d_in[3];
    const float* bl1   = (const float*)d_in[4];
    const float* Wr1   = (const float*)d_in[5];
    const float* Wl2   = (const float*)d_in[6];
    const float* bl2   = (const float*)d_in[7];
    const float* Wr2   = (const float*)d_in[8];

    const int N = in_sizes[0] / 64;   // 100000
    const int E = in_sizes[1] / 2;    // 1250000
    const int* src = ei;
    const int* dst = ei + E;

    // d_out layout: x (N*64 f32) | edge_index (2E i32 bits) | batch (N i32 bits)
    float* out_x     = (float*)d_out;
    int*   out_ei    = (int*)(out_x + (size_t)N * 64);
    int*   out_batch = out_ei + (size_t)2 * E;

    // workspace: agg (N*64 f32) + cnt (N f32)
    float* agg = (float*)d_ws;
    float* cnt = agg + (size_t)N * 64;

    const int TPB = 256;
    const int nAgg   = N * 64;
    const int nZero  = nAgg + N;                       // agg + cnt together
    const int gZero  = (nZero + TPB - 1) / TPB;
    const int gZeroA = (nAgg + TPB - 1) / TPB;
    const int gDeg   = (E + TPB - 1) / TPB;
    const int gScat  = (E * 16 + TPB - 1) / TPB;
    const int gGemm  = (N + 127) / 128;                // 8 waves x 16 rows / block

    // degrees (shared by both layers) + zero agg
    zero_f32_kernel<<<gZero, TPB, 0, stream>>>(agg, nZero);
    degree_kernel<<<gDeg, TPB, 0, stream>>>(dst, cnt, E);

    // layer 1: aggregate x -> agg, fused GEMM into out_x
    scatter_kernel<<<gScat, TPB, 0, stream>>>(x, src, dst, agg, E);
    sage_layer_kernel<<<gGemm, TPB, 0, stream>>>(x, agg, cnt, Wl1, bl1, Wr1, out_x, N);

    // layer 2: aggregate out_x -> agg, fused GEMM in place (per-(r,c) ownership)
    zero_f32_kernel<<<gZeroA, TPB, 0, stream>>>(agg, nAgg);
    scatter_kernel<<<gScat, TPB, 0, stream>>>(out_x, src, dst, agg, E);
    sage_layer_kernel<<<gGemm, TPB, 0, stream>>>(out_x, agg, cnt, Wl2, bl2, Wr2, out_x, N);

    // passthrough outputs
    copy_i32_kernel<<<(2 * E + TPB - 1) / TPB, TPB, 0, stream>>>(ei, out_ei, 2 * E);
    copy_i32_kernel<<<(N + TPB - 1) / TPB, TPB, 0, stream>>>(batch, out_batch, N);
}